// Job2vec_23751169147562
// MI455X (gfx1250) — compile-verified
//
#include <hip/hip_runtime.h>

typedef __attribute__((ext_vector_type(2))) float v2f;
typedef __attribute__((ext_vector_type(8))) float v8f;

#define VOCAB 100000
#define NDIM  128
#define BATCH 2048

#define ROW_TILES_PER_WAVE 2      // 32 rows
#define COL_TILES_PER_WAVE 5      // 80 cols
#define ROWS_PER_WAVE (ROW_TILES_PER_WAVE * 16)
#define COLS_PER_WAVE (COL_TILES_PER_WAVE * 16)
#define ROW_GROUPS (BATCH / ROWS_PER_WAVE)   // 64
#define COL_GROUPS (VOCAB / COLS_PER_WAVE)   // 1250
#define WAVES_PER_BLOCK 8
#define NUM_BLOCKS (ROW_GROUPS * COL_GROUPS / WAVES_PER_BLOCK)  // 10000

__global__ __launch_bounds__(256) void j2v_wmma_gemm(
    const int*   __restrict__ ids,   // [BATCH]
    const float* __restrict__ W1,    // [VOCAB, NDIM]
    const float* __restrict__ W2,    // [NDIM, VOCAB]
    float*       __restrict__ out)   // [BATCH, VOCAB]
{
    const int wave = blockIdx.x * WAVES_PER_BLOCK + (threadIdx.x >> 5);
    // consecutive waves in a block -> consecutive row groups, SAME col group,
    // so the 8 waves of a block reuse the same W2 strip out of WGP$/L2.
    const int rg = wave % ROW_GROUPS;
    const int cg = wave / ROW_GROUPS;

    const int lane = threadIdx.x & 31;
    const int half = lane >> 4;      // 0: lanes 0-15, 1: lanes 16-31
    const int l16  = lane & 15;

    const int row_base = rg * ROWS_PER_WAVE;
    const int col_base = cg * COLS_PER_WAVE;

    // Embedding gather: 16 row ids per row-tile (both lane halves need the
    // same 16 rows, indexed by l16).
    const int id0 = ids[row_base + l16];
    const int id1 = ids[row_base + 16 + l16];
    const float* __restrict__ A0 = W1 + (size_t)id0 * NDIM;
    const float* __restrict__ A1 = W1 + (size_t)id1 * NDIM;

    v8f acc[ROW_TILES_PER_WAVE][COL_TILES_PER_WAVE];
#pragma unroll
    for (int r = 0; r < ROW_TILES_PER_WAVE; ++r)
#pragma unroll
        for (int t = 0; t < COL_TILES_PER_WAVE; ++t)
            acc[r][t] = (v8f){0.f, 0.f, 0.f, 0.f, 0.f, 0.f, 0.f, 0.f};

    for (int k = 0; k < NDIM; k += 4) {
        const int kk = k + 2 * half;   // lane half selects K pair {k,k+1} / {k+2,k+3}

        // A fragments: 16x4 f32, VGPR0 = K=kk, VGPR1 = K=kk+1 (8B aligned -> b64 load)
        const v2f a0 = *(const v2f*)(A0 + kk);
        const v2f a1 = *(const v2f*)(A1 + kk);

        const float* __restrict__ Brow0 = W2 + (size_t)kk * VOCAB + col_base + l16;
        const float* __restrict__ Brow1 = Brow0 + VOCAB;

#pragma unroll
        for (int t = 0; t < COL_TILES_PER_WAVE; ++t) {
            // B fragment 4x16: row kk striped across lanes (half-split matches A)
            v2f b;
            b.x = Brow0[t * 16];
            b.y = Brow1[t * 16];
            acc[0][t] = __builtin_amdgcn_wmma_f32_16x16x4_f32(
                false, a0, false, b, (short)0, acc[0][t], false, false);
            acc[1][t] = __builtin_amdgcn_wmma_f32_16x16x4_f32(
                false, a1, false, b, (short)0, acc[1][t], false, false);
        }
    }

    // D layout: VGPR v holds row (v + 8*half) of the 16x16 tile, col = l16.
    // Non-temporal: 819 MB output stream must not evict W2 from the 192 MB L2.
#pragma unroll
    for (int r = 0; r < ROW_TILES_PER_WAVE; ++r) {
#pragma unroll
        for (int t = 0; t < COL_TILES_PER_WAVE; ++t) {
#pragma unroll
            for (int v = 0; v < 8; ++v) {
                const int row = row_base + r * 16 + v + 8 * half;
                const size_t off = (size_t)row * VOCAB + col_base + t * 16 + l16;
                __builtin_nontemporal_store(acc[r][t][v], out + off);
            }
        }
    }
}

extern "C" void kernel_launch(void* const* d_in, const int* in_sizes, int n_in,
                              void* d_out, int out_size, void* d_ws, size_t ws_size,
                              hipStream_t stream) {
    const int*   ids = (const int*)d_in[0];
    const float* W1  = (const float*)d_in[1];
    const float* W2  = (const float*)d_in[2];
    float*       out = (float*)d_out;

    j2v_wmma_gemm<<<dim3(NUM_BLOCKS), dim3(256), 0, stream>>>(ids, W1, W2, out);
}